// IDMPNN_full_39290360824129
// MI455X (gfx1250) — compile-verified
//
#include <hip/hip_runtime.h>
#include <math.h>

typedef __attribute__((ext_vector_type(2))) float v2f;
typedef __attribute__((ext_vector_type(8))) float v8f;

#define H       64
#define NGRAPH  16
#define NNODE   64
#define SUBG_PER 128
#define NS      2048
#define KSUB    4
#define PPERM   6
#define NDEG    65
#define NMASK   16
#define NBASE   5
#define NCOMBO  (NBASE*NDEG*NMASK)   /* 5200 */
#define VROWT   (NCOMBO/16)          /* 325  */

__device__ __forceinline__ float selu_f(float x) {
  const float scale = 1.0507009873554805f;
  const float alpha = 1.6732632423543772f;
  return x > 0.0f ? scale * x : scale * alpha * (expf(x) - 1.0f);
}

// ---------------------------------------------------------------------------
// Precompute: lab, labsum, colsum(id_w), TW[mask][p][:] = T[mask][p][:] @ id_w
// ---------------------------------------------------------------------------
__global__ void precompute1_kernel(const float* __restrict__ idemb,
                                   const float* __restrict__ id_w,
                                   float* __restrict__ labsum,
                                   float* __restrict__ TW,
                                   float* __restrict__ wcs) {
  __shared__ float labL[KSUB * PPERM * H];     // lab[j][p][d], 6 KB
  // itertools.permutations(range(1,4)) order, with 0 prepended:
  const int permtab[PPERM][KSUB] = {
      {0,1,2,3},{0,1,3,2},{0,2,1,3},{0,2,3,1},{0,3,1,2},{0,3,2,1}};
  int tid = threadIdx.x;                       // 256 threads
  for (int idx = tid; idx < KSUB*PPERM*H; idx += blockDim.x) {
    int j = idx / (PPERM*H);
    int p = (idx / H) % PPERM;
    int d = idx & (H-1);
    labL[idx] = idemb[permtab[p][j]*H + d] * ((float)(j+1) * 1000.0f);
  }
  __syncthreads();
  for (int idx = tid; idx < KSUB*H; idx += blockDim.x) {   // labsum[j][d]
    int j = idx / H, d = idx & (H-1);
    float s = 0.f;
    for (int p = 0; p < PPERM; p++) s += labL[(j*PPERM + p)*H + d];
    labsum[idx] = s;
  }
  for (int dp = tid; dp < H; dp += blockDim.x) {           // column sums of id_w
    float s = 0.f;
    for (int d = 0; d < H; d++) s += id_w[d*H + dp];
    wcs[dp] = s;
  }
  for (int idx = tid; idx < NMASK*PPERM*H; idx += blockDim.x) {
    int mask = idx / (PPERM*H);
    int p = (idx / H) % PPERM;
    int dp = idx & (H-1);
    float acc = 0.f;
    for (int d = 0; d < H; d++) {
      float t = 0.f;
      if (mask & 1) t += labL[(0*PPERM + p)*H + d] - 1.0f;
      if (mask & 2) t += labL[(1*PPERM + p)*H + d] - 1.0f;
      if (mask & 4) t += labL[(2*PPERM + p)*H + d] - 1.0f;
      if (mask & 8) t += labL[(3*PPERM + p)*H + d] - 1.0f;
      acc += t * id_w[d*H + dp];
    }
    TW[idx] = acc;
  }
}

// ---------------------------------------------------------------------------
// Row degrees: rowdeg[g,i] = sum_j adj[g,i,j]   (exact small-integer floats)
// ---------------------------------------------------------------------------
__global__ void rowdeg_kernel(const float* __restrict__ adj,
                              float* __restrict__ rowdeg) {
  int g = blockIdx.x, i = threadIdx.x;
  const float* r = adj + (g*NNODE + i)*NNODE;
  float s = 0.f;
  for (int j = 0; j < NNODE; j++) s += r[j];
  rowdeg[g*NNODE + i] = s;
}

// ---------------------------------------------------------------------------
// SelSum[deg,mask,d] = sum_p selu(deg*wcs[d] + TW[mask][p][d] + id_b[d])
// ---------------------------------------------------------------------------
__global__ void selsum_kernel(const float* __restrict__ wcs,
                              const float* __restrict__ TW,
                              const float* __restrict__ id_b,
                              float* __restrict__ SelSum) {
  int b = blockIdx.x;               // b = deg*NMASK + mask
  int deg = b / NMASK, mask = b % NMASK;
  int d = threadIdx.x;
  float base = (float)deg * wcs[d] + id_b[d];
  float acc = 0.f;
  for (int p = 0; p < PPERM; p++)
    acc += selu_f(base + TW[(mask*PPERM + p)*H + d]);
  SelSum[b*H + d] = acc;
}

// ---------------------------------------------------------------------------
// 16-row f32 GEMM tile via V_WMMA_F32_16X16X4_F32: out = selu(Ut @ W + bias)
// Block = 128 threads (4 waves), wave w computes N-tile w.
// A layout (16x4 f32): a[v] = A[m, k0 + 2*hi + v];  B symmetric;
// C/D (16x16 f32): VGPR r holds row r+8*hi, col = lane&15 (+16*ntile).
// ---------------------------------------------------------------------------
__device__ __forceinline__ void gemm16_selu(const float* Ut,          // LDS 16xH
                                            const float* __restrict__ W,
                                            const float* __restrict__ bias,
                                            float* __restrict__ outp,
                                            int rowbase) {
  int tid  = threadIdx.x;
  int wave = tid >> 5;
  int lane = tid & 31;
  int m    = lane & 15;
  int hi   = lane >> 4;
  int n    = wave*16 + m;
  v8f acc = {0.f,0.f,0.f,0.f,0.f,0.f,0.f,0.f};
  #pragma unroll
  for (int k0 = 0; k0 < H; k0 += 4) {
    int ka = k0 + 2*hi;
    v2f a, b;
    a.x = Ut[m*H + ka];
    a.y = Ut[m*H + ka + 1];
    b.x = W[ka*H + n];
    b.y = W[(ka + 1)*H + n];
    acc = __builtin_amdgcn_wmma_f32_16x16x4_f32(false, a, false, b,
                                                (short)0, acc, false, false);
  }
  float bv = bias[n];
  #pragma unroll
  for (int r = 0; r < 8; r++) {
    int row = rowbase + r + 8*hi;
    outp[row*H + n] = selu_f(acc[r] + bv);
  }
}

// V[combo,:] = selu(U[combo,:] @ set1_w + set1_b), combo=(base*65+deg)*16+mask
__global__ void v_table_kernel(const float* __restrict__ labsum,
                               const float* __restrict__ SelSum,
                               const float* __restrict__ set1_w,
                               const float* __restrict__ set1_b,
                               float* __restrict__ V) {
  __shared__ float Ut[16 * H];
  int rt = blockIdx.x;                          // 0..324
  int tid = threadIdx.x;                        // 128
  for (int idx = tid; idx < 16*H; idx += 128) {
    int rrow = idx >> 6;
    int k    = idx & 63;
    int c    = rt*16 + rrow;
    int mask = c & 15;
    int dm   = c >> 4;                          // base*65 + deg
    int deg  = dm % NDEG;
    int base = dm / NDEG;
    float bv = (base == 0) ? 6.0f : labsum[(base-1)*H + k];
    Ut[idx] = bv + SelSum[(deg*NMASK + mask)*H + k];
  }
  __syncthreads();
  gemm16_selu(Ut, set1_w, set1_b, V, rt*16);
}

// ---------------------------------------------------------------------------
// Per subgraph: build combo index per node, then g0[s,:] = sum_i V[combo_i,:]
// ---------------------------------------------------------------------------
__global__ void combo_sum_kernel(const float* __restrict__ adj,
                                 const int* __restrict__ subgs,
                                 const float* __restrict__ rowdeg,
                                 const float* __restrict__ V,
                                 float* __restrict__ g0) {
  __shared__ int comboL[NNODE];
  int s = blockIdx.x;
  int tid = threadIdx.x;                        // 64
  int g = s >> 7;                               // num_subg == 128 per graph
  int c0 = subgs[s*4+0], c1 = subgs[s*4+1], c2 = subgs[s*4+2], c3 = subgs[s*4+3];
  {
    int i = tid;
    const float* arow = adj + (g*NNODE + i)*NNODE;
    __builtin_prefetch(arow, 0, 3);
    int mask = (arow[c0] > 0.5f ? 1 : 0) | (arow[c1] > 0.5f ? 2 : 0)
             | (arow[c2] > 0.5f ? 4 : 0) | (arow[c3] > 0.5f ? 8 : 0);
    int deg  = (int)(rowdeg[g*NNODE + i] + 0.5f);
    int base = (i == c0) ? 1 : (i == c1) ? 2 : (i == c2) ? 3 : (i == c3) ? 4 : 0;
    comboL[i] = (base*NDEG + deg)*NMASK + mask;
  }
  __syncthreads();
  float acc = 0.f;
  for (int i = 0; i < NNODE; i++) acc += V[comboL[i]*H + tid];
  g0[s*H + tid] = acc;
}

// g_arr[s,:] = selu(g0[s,:] @ set2_w + set2_b)  (WMMA, 128 row tiles)
__global__ void set2_kernel(const float* __restrict__ g0,
                            const float* __restrict__ set2_w,
                            const float* __restrict__ set2_b,
                            float* __restrict__ g_arr) {
  __shared__ float Ut[16 * H];
  int rt = blockIdx.x;
  int tid = threadIdx.x;
  for (int idx = tid; idx < 16*H; idx += 128) Ut[idx] = g0[rt*16*H + idx];
  __syncthreads();
  gemm16_selu(Ut, set2_w, set2_b, g_arr, rt*16);
}

// ---------------------------------------------------------------------------
// Per graph: segment-max over 128 subgraphs, 4 residual MLP+LN blocks, output
// ---------------------------------------------------------------------------
__global__ void head_kernel(const float* __restrict__ g_arr,
                            const float* __restrict__ rw1, const float* __restrict__ rb1,
                            const float* __restrict__ rw2, const float* __restrict__ rb2,
                            const float* __restrict__ lng, const float* __restrict__ lnb,
                            const float* __restrict__ out_w, const float* __restrict__ out_b,
                            float* __restrict__ out) {
  __shared__ float row[H];
  __shared__ float t1[H];
  __shared__ float t2[H];
  __shared__ float red[H];
  int g = blockIdx.x;
  int d = threadIdx.x;                          // 64
  float v = -3.402823466e38f;
  for (int s = 0; s < SUBG_PER; s++)
    v = fmaxf(v, g_arr[(g*SUBG_PER + s)*H + d]);
  row[d] = v;
  __syncthreads();
  for (int L = 0; L < 4; L++) {
    const float* W1 = rw1 + L*H*H;  const float* B1 = rb1 + L*H;
    const float* W2 = rw2 + L*H*H;  const float* B2 = rb2 + L*H;
    float a = B1[d];
    for (int k = 0; k < H; k++) a += row[k] * W1[k*H + d];
    t1[d] = fmaxf(a, 0.f);
    __syncthreads();
    float b = B2[d];
    for (int k = 0; k < H; k++) b += t1[k] * W2[k*H + d];
    t2[d] = b;
    __syncthreads();
    float mean = 0.f;
    for (int k = 0; k < H; k++) mean += t2[k];
    mean *= (1.0f / H);
    float var = 0.f;
    for (int k = 0; k < H; k++) { float dd = t2[k] - mean; var += dd*dd; }
    var *= (1.0f / H);
    float ln = (b - mean) * rsqrtf(var + 1e-5f) * lng[L*H + d] + lnb[L*H + d];
    float t  = fmaxf(ln, 0.f);
    __syncthreads();
    row[d] = row[d] + t;
    __syncthreads();
  }
  red[d] = row[d] * out_w[d];
  __syncthreads();
  if (d == 0) {
    float acc = out_b[0];
    for (int k = 0; k < H; k++) acc += red[k];
    out[g] = acc;
  }
}

// ---------------------------------------------------------------------------
extern "C" void kernel_launch(void* const* d_in, const int* in_sizes, int n_in,
                              void* d_out, int out_size, void* d_ws, size_t ws_size,
                              hipStream_t stream) {
  (void)in_sizes; (void)n_in; (void)out_size; (void)ws_size;
  const float* adj    = (const float*)d_in[1];
  const int*   subgs  = (const int*)  d_in[2];
  const float* idemb  = (const float*)d_in[10];
  const float* id_w   = (const float*)d_in[11];
  const float* id_b   = (const float*)d_in[12];
  const float* set1_w = (const float*)d_in[13];
  const float* set1_b = (const float*)d_in[14];
  const float* set2_w = (const float*)d_in[15];
  const float* set2_b = (const float*)d_in[16];
  const float* reg_w1 = (const float*)d_in[17];
  const float* reg_b1 = (const float*)d_in[18];
  const float* reg_w2 = (const float*)d_in[19];
  const float* reg_b2 = (const float*)d_in[20];
  const float* lng    = (const float*)d_in[21];
  const float* lnb    = (const float*)d_in[22];
  const float* out_w  = (const float*)d_in[23];
  const float* out_b  = (const float*)d_in[24];
  float* out = (float*)d_out;

  float* ws      = (float*)d_ws;
  float* labsum  = ws;                 // 4*64        = 256
  float* TW      = labsum + 256;       // 16*6*64     = 6144
  float* wcs     = TW + 6144;          // 64
  float* rowdeg  = wcs + 64;           // 16*64       = 1024
  float* SelSum  = rowdeg + 1024;      // 65*16*64    = 66560
  float* V       = SelSum + 66560;     // 5200*64     = 332800
  float* g0      = V + 332800;         // 2048*64     = 131072
  float* g_arr   = g0 + 131072;        // 2048*64     = 131072  (total ~2.7 MB)

  hipLaunchKernelGGL(precompute1_kernel, dim3(1),    dim3(256), 0, stream,
                     idemb, id_w, labsum, TW, wcs);
  hipLaunchKernelGGL(rowdeg_kernel,      dim3(NGRAPH), dim3(NNODE), 0, stream,
                     adj, rowdeg);
  hipLaunchKernelGGL(selsum_kernel,      dim3(NDEG*NMASK), dim3(H), 0, stream,
                     wcs, TW, id_b, SelSum);
  hipLaunchKernelGGL(v_table_kernel,     dim3(VROWT), dim3(128), 0, stream,
                     labsum, SelSum, set1_w, set1_b, V);
  hipLaunchKernelGGL(combo_sum_kernel,   dim3(NS),   dim3(NNODE), 0, stream,
                     adj, subgs, rowdeg, V, g0);
  hipLaunchKernelGGL(set2_kernel,        dim3(NS/16), dim3(128), 0, stream,
                     g0, set2_w, set2_b, g_arr);
  hipLaunchKernelGGL(head_kernel,        dim3(NGRAPH), dim3(H), 0, stream,
                     g_arr, reg_w1, reg_b1, reg_w2, reg_b2, lng, lnb, out_w, out_b, out);
}